// NKQuantizer2_33389075759172
// MI455X (gfx1250) — compile-verified
//
#include <hip/hip_runtime.h>
#include <float.h>
#include <stdint.h>

// Problem constants from the reference: x [B*S, Q], W [E, Q], K=8 top codes.
#define QDIM     8192
#define TOPK     8
#define TPB_TOPK 256
#define CHUNKS   8      // 8 chunks * 256 threads * 4 floats = 8192 = QDIM

typedef __attribute__((ext_vector_type(2))) float v2f;
typedef __attribute__((ext_vector_type(8))) float v8f;
typedef __attribute__((ext_vector_type(4))) int   v4i;

// Pointer-to-v4i in global (AS1) and LDS (AS3) address spaces, as required by
// __builtin_amdgcn_global_load_async_to_lds_b128 (per hipcc diagnostic).
typedef __attribute__((address_space(1))) v4i* gv4i_p;
typedef __attribute__((address_space(3))) v4i* lv4i_p;

// ---------------------------------------------------------------------------
// CDNA5 async global->LDS copy (16B per lane), ASYNCcnt-tracked.
// ---------------------------------------------------------------------------
__device__ __forceinline__ void async_copy16(void* lds_dst, const void* gsrc) {
#if __has_builtin(__builtin_amdgcn_global_load_async_to_lds_b128)
  __builtin_amdgcn_global_load_async_to_lds_b128(
      (gv4i_p)(uintptr_t)gsrc,      // global source (flat -> AS1, same bits)
      (lv4i_p)(uintptr_t)lds_dst,   // LDS dest (low 32 bits = LDS offset)
      /*imm offset*/0, /*cpol*/0);
#else
  // ISA 15.18.3: GLOBAL_LOAD_ASYNC_TO_LDS_B128  vdst(LDS byte addr), vaddr, off
  unsigned loff = (unsigned)(uintptr_t)lds_dst;
  asm volatile("global_load_async_to_lds_b128 %0, %1, off"
               :: "v"(loff), "v"(gsrc) : "memory");
#endif
}

__device__ __forceinline__ void wait_async_zero() {
#if __has_builtin(__builtin_amdgcn_s_wait_asynccnt)
  __builtin_amdgcn_s_wait_asynccnt(0);
  asm volatile("" ::: "memory");
#else
  asm volatile("s_wait_asynccnt 0x0" ::: "memory");
#endif
}

// ---------------------------------------------------------------------------
// Kernel 0: transpose W [E, Q] -> Wt [Q, E] (makes the gather coalesced).
// ---------------------------------------------------------------------------
__global__ void __launch_bounds__(256)
nk_transpose_kernel(const float* __restrict__ W, float* __restrict__ Wt,
                    int E, int Q) {
  __shared__ float tile[32][33];
  const int tx = threadIdx.x, ty = threadIdx.y;
  const int qt = blockIdx.x * 32, et = blockIdx.y * 32;
  #pragma unroll
  for (int i = ty; i < 32; i += 8)
    tile[i][tx] = W[(size_t)(et + i) * Q + (qt + tx)];
  __syncthreads();
  #pragma unroll
  for (int i = ty; i < 32; i += 8)
    Wt[(size_t)(qt + i) * E + (et + tx)] = tile[tx][i];
}

// ---------------------------------------------------------------------------
// Kernel 1: per-token top-8 (one block per token). x row streamed to LDS via
// async-to-LDS, held in registers, 8 rounds of block-wide argmax with
// lower-index tiebreak (matches jax.lax.top_k ordering semantics).
// If idx_out==nullptr: fused fallback, gathers directly from W (no workspace).
// ---------------------------------------------------------------------------
__global__ void __launch_bounds__(TPB_TOPK)
nk_topk_kernel(const float* __restrict__ x, const float* __restrict__ W,
               int* __restrict__ idx_out, float* __restrict__ out, int E) {
  __shared__ float xs[QDIM];            // 32 KB staging for one token row
  __shared__ float sval[TPB_TOPK];
  __shared__ int   sidx[TPB_TOPK];
  __shared__ int   win[TOPK];

  const int tid = threadIdx.x;
  const size_t token = blockIdx.x;
  const float* row = x + token * (size_t)QDIM;

  // Stream the row into LDS with CDNA5 async loads (coalesced 16B/lane).
  #pragma unroll
  for (int c = 0; c < CHUNKS; ++c) {
    const int q4 = c * (TPB_TOPK * 4) + tid * 4;
    async_copy16(&xs[q4], row + q4);
  }
  wait_async_zero();   // each lane only reads back the bytes it wrote

  // Register-resident copy: 32 floats/thread -> 8 scan rounds are free.
  float r[32];
  #pragma unroll
  for (int c = 0; c < CHUNKS; ++c) {
    float4 v = *reinterpret_cast<const float4*>(&xs[c * (TPB_TOPK * 4) + tid * 4]);
    r[c * 4 + 0] = v.x; r[c * 4 + 1] = v.y;
    r[c * 4 + 2] = v.z; r[c * 4 + 3] = v.w;
  }

  #pragma unroll 1
  for (int round = 0; round < TOPK; ++round) {
    // Local argmax (ascending q order => strict '>' keeps lowest index on tie).
    float lv = -FLT_MAX; int li = 0x7fffffff;
    #pragma unroll
    for (int k = 0; k < 32; ++k) {
      const int q = (k >> 2) * (TPB_TOPK * 4) + tid * 4 + (k & 3);
      if (r[k] > lv) { lv = r[k]; li = q; }
    }
    sval[tid] = lv; sidx[tid] = li;
    __syncthreads();
    #pragma unroll
    for (int s = TPB_TOPK / 2; s > 0; s >>= 1) {
      if (tid < s) {
        const float v1 = sval[tid], v2 = sval[tid + s];
        const int   i1 = sidx[tid], i2 = sidx[tid + s];
        if (v2 > v1 || (v2 == v1 && i2 < i1)) { sval[tid] = v2; sidx[tid] = i2; }
      }
      __syncthreads();
    }
    const int widx = sidx[0];
    if (tid == 0) win[round] = widx;
    __syncthreads();
    // Owner knocks the winner out for subsequent rounds (no dynamic indexing).
    #pragma unroll
    for (int k = 0; k < 32; ++k) {
      const int q = (k >> 2) * (TPB_TOPK * 4) + tid * 4 + (k & 3);
      if (q == widx) r[k] = -FLT_MAX;
    }
  }

  if (idx_out != nullptr) {
    if (tid < TOPK) idx_out[token * TOPK + tid] = win[tid];
  } else {
    // Fused fallback: gather columns straight from W [E, Q] (L2-resident).
    for (int e = tid; e < E; e += TPB_TOPK) {
      float acc = 0.f;
      #pragma unroll
      for (int j = 0; j < TOPK; ++j) acc += W[(size_t)e * QDIM + win[j]];
      out[token * (size_t)E + e] = acc;
    }
  }
}

// ---------------------------------------------------------------------------
// Kernel 2: gather-sum via V_WMMA_F32_16X16X4_F32 (exact fp32).
// A = all-ones 16x4  =>  D[m,n] = sum_k B[k,n]  for every m (layout-robust:
// summation over k is symmetric, lane -> n is forced, C/D layout documented).
// B rows = 4 gathered Wt rows; two chained WMMAs accumulate all 8 rows.
// One wave per token; 32 column-tiles of 16 e-values each.
// ---------------------------------------------------------------------------
__global__ void __launch_bounds__(128)
nk_gather_wmma_kernel(const float* __restrict__ Wt,
                      const int* __restrict__ idxbuf,
                      float* __restrict__ out, int E) {
  const int lane = threadIdx.x & 31;
  const int wave = threadIdx.x >> 5;
  const size_t token = (size_t)blockIdx.x * 4 + wave;
  const int* ip = idxbuf + token * TOPK;
  const int i0 = ip[0], i1 = ip[1], i2 = ip[2], i3 = ip[3];
  const int i4 = ip[4], i5 = ip[5], i6 = ip[6], i7 = ip[7];

  const int  ln = lane & 15;
  const bool hi = lane >= 16;
  // VGPR0 lanes0-15 = row K0, lanes16-31 = row K2; VGPR1 = rows K1, K3.
  const float* r0 = Wt + (size_t)(hi ? i2 : i0) * E;
  const float* r1 = Wt + (size_t)(hi ? i3 : i1) * E;
  const float* r2 = Wt + (size_t)(hi ? i6 : i4) * E;
  const float* r3 = Wt + (size_t)(hi ? i7 : i5) * E;
  float* orow = out + token * (size_t)E;

  v2f a; a[0] = 1.0f; a[1] = 1.0f;     // all-ones A matrix

  for (int e0 = 0; e0 < E; e0 += 16) {
    v2f b0, b1;
    b0[0] = r0[e0 + ln]; b0[1] = r1[e0 + ln];
    b1[0] = r2[e0 + ln]; b1[1] = r3[e0 + ln];
    v8f c = {};
#if __has_builtin(__builtin_amdgcn_wmma_f32_16x16x4_f32)
    c = __builtin_amdgcn_wmma_f32_16x16x4_f32(false, a, false, b0,
                                              (short)0, c, false, false);
    c = __builtin_amdgcn_wmma_f32_16x16x4_f32(false, a, false, b1,
                                              (short)0, c, false, false);
    if (!hi) orow[e0 + ln] = c[0];     // D[0, n] = sum of the 8 gathered rows
#else
    // Scalar fallback (exact same math) if the builtin is unavailable.
    if (!hi) {
      float s0 = r0[e0 + ln] + r1[e0 + ln] + r2[e0 + ln] + r3[e0 + ln];
      const float* q0 = Wt + (size_t)i2 * E; const float* q1 = Wt + (size_t)i3 * E;
      const float* q2 = Wt + (size_t)i6 * E; const float* q3 = Wt + (size_t)i7 * E;
      s0 += q0[e0 + ln] + q1[e0 + ln] + q2[e0 + ln] + q3[e0 + ln];
      orow[e0 + ln] = s0;
    }
#endif
  }
}

// ---------------------------------------------------------------------------
extern "C" void kernel_launch(void* const* d_in, const int* in_sizes, int n_in,
                              void* d_out, int out_size, void* d_ws, size_t ws_size,
                              hipStream_t stream) {
  (void)n_in; (void)out_size;
  const int Q = QDIM;
  const int N = in_sizes[0] / Q;   // B*S tokens (8192)
  const int E = in_sizes[1] / Q;   // embed dim (512)

  const float* x = (const float*)d_in[0];
  const float* W = (const float*)d_in[1];
  float* out = (float*)d_out;

  const size_t wt_bytes  = (size_t)Q * (size_t)E * sizeof(float);
  const size_t idx_bytes = (size_t)N * TOPK * sizeof(int);

  if (ws_size >= wt_bytes + idx_bytes) {
    float* Wt     = (float*)d_ws;
    int*   idxbuf = (int*)((char*)d_ws + wt_bytes);
    hipLaunchKernelGGL(nk_transpose_kernel, dim3(Q / 32, E / 32), dim3(32, 8),
                       0, stream, W, Wt, E, Q);
    hipLaunchKernelGGL(nk_topk_kernel, dim3(N), dim3(TPB_TOPK),
                       0, stream, x, W, idxbuf, (float*)nullptr, E);
    hipLaunchKernelGGL(nk_gather_wmma_kernel, dim3(N / 4), dim3(128),
                       0, stream, Wt, idxbuf, out, E);
  } else {
    // Workspace too small: fused top-k + direct gather (no scratch needed).
    hipLaunchKernelGGL(nk_topk_kernel, dim3(N), dim3(TPB_TOPK),
                       0, stream, x, W, (int*)nullptr, out, E);
  }
}